// Eikonal_10943576670376
// MI455X (gfx1250) — compile-verified
//
#include <hip/hip_runtime.h>
#include <hip/hip_bf16.h>
#include <math.h>

// ---------------------------------------------------------------------------
// Eikonal graph update for MI455X (gfx1250, wave32).
//   deg  = segment_sum(edge_attr, src)                         [N]
//   gm   = sqrt(w) * max((y[b,src]-y[b,dst])/deg[src], 0)      [B,E]
//   gn   = segment_max(gm, src)                                [B,N]
//   out  = (1 - gn) * mask                                     [B,N]
//
// Memory/atomic-bound scatter-gather (AI ~0.5 flop/byte): WMMA inapplicable.
// CDNA5 levers used instead:
//  * y transposed to [N,8]: 8 lanes (one edge x 8 batches) gather one
//    coalesced 32B line from L2 (y_t, deg, gn all L2-resident; 192MB L2).
//  * Tensor Data Mover stages the edge stream into double-buffered LDS
//    (tensor_load_to_lds + s_wait_tensorcnt): frees the vector-load path,
//    and the 8-lanes-per-edge reads become conflict-free LDS broadcasts.
//  * Non-returning atomics at the L2 units: global_atomic_add_f32 for the
//    degree sum; gm >= 0 so unsigned bit-pattern atomic max is exact.
// ---------------------------------------------------------------------------

#define BATCH 8
#define TILE  1024          // edges per TDM chunk
#define CPW   2             // chunks per workgroup (double-buffered)

typedef unsigned int v4u __attribute__((ext_vector_type(4)));
typedef int          v8i __attribute__((ext_vector_type(8)));
typedef int          v4i __attribute__((ext_vector_type(4)));

// ---- Kernel A: zero accumulators + transpose y [B,N] -> yt [N,B].
// 2D grid (blockIdx.y = batch) -> no integer division.
__global__ void eik_init_kernel(const float* __restrict__ y,
                                float* __restrict__ yt,   // [N*B]
                                float* __restrict__ gn,   // [N*B] (zeroed)
                                float* __restrict__ deg,  // [N]   (zeroed)
                                int N) {
    int n = blockIdx.x * blockDim.x + threadIdx.x;
    int b = blockIdx.y;
    if (n >= N) return;
    long long t = (long long)b * N + n;       // coalesced read of y
    yt[(long long)n * BATCH + b] = y[t];
    gn[t] = 0.0f;                             // any full cover of N*B works
    if (b == 0) deg[n] = 0.0f;
}

// ---- Kernel B: weighted out-degree via scatter-add (global_atomic_add_f32).
__global__ void eik_degree_kernel(const long long* __restrict__ ei, // [2*E]
                                  const float* __restrict__ ea,     // [E]
                                  float* __restrict__ deg,          // [N]
                                  int E) {
    int e = blockIdx.x * blockDim.x + threadIdx.x;
    if (e >= E) return;
    __builtin_prefetch(ei + e + 2048, 0, 0);  // global_prefetch_b8
    __builtin_prefetch(ea + e + 2048, 0, 0);
    long long s = ei[e];
    atomicAdd(&deg[s], ea[e]);
}

#if defined(__HIP_DEVICE_COMPILE__) && __has_builtin(__builtin_amdgcn_tensor_load_to_lds)
#define EIK_HAS_TDM 1

// Issue one TDM load: `rows` rows of TILE elements, row stride `stride0`
// elements, element size 2^dsz_log2 bytes, OOB (x >= tdim0) zero-fills.
// D# layout per CDNA5 ISA.
__device__ __forceinline__ void tdm_issue(const void* gbase, unsigned lds_off,
                                          unsigned rows,
                                          unsigned long long tdim0,
                                          unsigned long long stride0,
                                          int dsz_log2) {
    unsigned long long ga = (unsigned long long)gbase;
    v4u g0;
    g0.x = 1u;                                            // count=1, user mode
    g0.y = lds_off;                                       // lds_addr [63:32]
    g0.z = (unsigned)ga;                                  // global_addr lo
    g0.w = (unsigned)((ga >> 32) & 0x01FFFFFFu) | (2u << 30); // addr hi | type=2
    unsigned td0 = (tdim0 > 0xFFFFFFFFull) ? 0xFFFFFFFFu : (unsigned)tdim0;
    v8i g1;
    g1[0] = (int)((unsigned)dsz_log2 << 16);              // wg_mask=0, data_size
    g1[1] = (int)((td0 & 0xFFFFu) << 16);                 // tensor_dim0 lo16
    g1[2] = (int)(((td0 >> 16) & 0xFFFFu) | (rows << 16)); // td0 hi16 | tensor_dim1
    g1[3] = (int)((unsigned)TILE << 16);                  // tile_dim0
    g1[4] = (int)rows;                                    // tile_dim1 (tile_dim2=0)
    g1[5] = (int)(unsigned)(stride0 & 0xFFFFFFFFull);     // dim0 stride lo32
    g1[6] = (int)(unsigned)((stride0 >> 32) & 0xFFFFull); // dim0 stride hi16
    g1[7] = 0;                                            // dim1 stride (unused)
    v4i z4 = {0, 0, 0, 0};
#if __clang_major__ >= 23
    v8i z8 = {0, 0, 0, 0, 0, 0, 0, 0};
    __builtin_amdgcn_tensor_load_to_lds(g0, g1, z4, z4, z8, 0);
#else
    __builtin_amdgcn_tensor_load_to_lds(g0, g1, z4, z4, 0);
#endif
}

__device__ __forceinline__ unsigned lds_off32(const void* p) {
    // flat pointer to LDS: aperture in [63:32], byte offset in [31:0]
    return (unsigned)(unsigned long long)p;
}

// 2 TDM ops per chunk: (a) both index rows (src row + dst row of edge_index,
// row stride E) in one 2-row descriptor; (b) the edge_attr row.
__device__ __forceinline__ void issue_chunk(const long long* ei, const float* ea,
                                            int E, int base,
                                            long long* lds_idx, float* lds_w) {
    unsigned long long rem = (unsigned long long)(E - base);
    tdm_issue(ei + base, lds_off32(lds_idx), 2u, rem, (unsigned long long)E, 3);
    tdm_issue(ea + base, lds_off32(lds_w),   1u, rem, rem,                   2);
}
#endif  // EIK_HAS_TDM

// ---- Kernel C: per-edge, per-batch message + scatter-max.
// Each workgroup owns CPW*TILE edges; wave 0 DMAs chunks into double-buffered
// LDS via the TDM, all 8 waves process 8 lanes/edge out of LDS.
__global__ void eik_edge_kernel(const long long* __restrict__ ei,  // [2*E]
                                const float* __restrict__ ea,      // [E]
                                const float* __restrict__ deg,     // [N]
                                const float* __restrict__ yt,      // [N*B]
                                unsigned int* __restrict__ gn_u,   // [N*B]
                                const int* __restrict__ alpha_p,
                                int E) {
    const int wg_base = blockIdx.x * (TILE * CPW);
    const int alpha = alpha_p[0];
#if defined(EIK_HAS_TDM)
    __shared__ long long s_idx[2][2 * TILE];   // [buf][src row | dst row]
    __shared__ float     s_w[2][TILE];

    int remaining = E - wg_base;
    if (remaining <= 0) return;
    int nc = (remaining + TILE - 1) / TILE;
    if (nc > CPW) nc = CPW;

    const bool issuer = (threadIdx.x < 32);    // wave 0 drives the TDM
    if (issuer) {
        issue_chunk(ei, ea, E, wg_base, &s_idx[0][0], &s_w[0][0]);
        if (nc > 1)
            issue_chunk(ei, ea, E, wg_base + TILE, &s_idx[1][0], &s_w[1][0]);
    }
    for (int k = 0; k < nc; ++k) {
        if (issuer) {
            // in-order completion: cnt<=2 means chunk k's 2 DMAs finished
            if (k + 1 < nc) __builtin_amdgcn_s_wait_tensorcnt(2);
            else            __builtin_amdgcn_s_wait_tensorcnt(0);
        }
        __syncthreads();                       // buf[k&1] valid for all waves
        const long long* csrc = &s_idx[k & 1][0];
        const long long* cdst = &s_idx[k & 1][TILE];
        const float*     cw   = &s_w[k & 1][0];
        const int base = wg_base + k * TILE;
        for (int i = threadIdx.x; i < TILE * BATCH; i += blockDim.x) {
            int el = i >> 3;                   // edge within chunk
            int b  = i & 7;                    // batch
            int e  = base + el;
            if (e < E) {
                long long s = csrc[el];        // LDS broadcast (8 lanes/addr)
                long long d = cdst[el];
                float w  = cw[el];
                float dg = deg[s];
                float denom = (alpha == 1) ? dg : __powf(dg, (float)alpha);
                float ys = yt[s * BATCH + b];  // 32B-coalesced L2 gather
                float yd = yt[d * BATCH + b];
                float gm = sqrtf(w) * fmaxf((ys - yd) / denom, 0.0f);
                atomicMax(&gn_u[s * BATCH + b], __float_as_uint(gm));
            }
        }
        __syncthreads();                       // done reading before buf reuse
        if (issuer && (k + CPW) < nc)          // general issue-ahead (nc<=CPW: no-op)
            issue_chunk(ei, ea, E, wg_base + (k + CPW) * TILE,
                        &s_idx[k & 1][0], &s_w[k & 1][0]);
    }
#else
    // Direct-load fallback (host pass / toolchains without the TDM builtin).
    for (int i = threadIdx.x; i < TILE * CPW * BATCH; i += blockDim.x) {
        int el = i >> 3;
        int b  = i & 7;
        int e  = wg_base + el;
        if (e < E) {
            long long s = ei[e];
            long long d = ei[(long long)E + e];
            float w  = ea[e];
            float dg = deg[s];
            float denom = (alpha == 1) ? dg : __powf(dg, (float)alpha);
            float ys = yt[s * BATCH + b];
            float yd = yt[d * BATCH + b];
            float gm = sqrtf(w) * fmaxf((ys - yd) / denom, 0.0f);
            atomicMax(&gn_u[s * BATCH + b], __float_as_uint(gm));
        }
    }
#endif
}

// ---- Kernel D: out[b,n] = (1 - max(gn,0)) * mask[b,n]  (back to [B,N]).
__global__ void eik_final_kernel(const float* __restrict__ gn,   // [N*B]
                                 const float* __restrict__ mask, // [B*N]
                                 float* __restrict__ out,        // [B*N]
                                 int N) {
    int n = blockIdx.x * blockDim.x + threadIdx.x;
    int b = blockIdx.y;
    if (n >= N) return;
    long long t = (long long)b * N + n;
    float g = gn[(long long)n * BATCH + b];
    out[t] = (1.0f - fmaxf(g, 0.0f)) * mask[t];
}

extern "C" void kernel_launch(void* const* d_in, const int* in_sizes, int n_in,
                              void* d_out, int out_size, void* d_ws, size_t ws_size,
                              hipStream_t stream) {
    // inputs (setup_inputs order): t, y, edge_index, edge_attr, mask, alpha
    const float*     y     = (const float*)d_in[1];
    const long long* ei    = (const long long*)d_in[2];   // int64 [2,E]
    const float*     ea    = (const float*)d_in[3];
    const float*     mask  = (const float*)d_in[4];
    const int*       alpha = (const int*)d_in[5];
    float* out = (float*)d_out;

    const int BN = in_sizes[1];        // B*N
    const int N  = BN / BATCH;
    const int E  = in_sizes[3];

    // workspace layout: yt [N*B] | gn [N*B] | deg [N]
    float* yt  = (float*)d_ws;
    float* gn  = yt + (size_t)N * BATCH;
    float* deg = gn + (size_t)N * BATCH;

    const int TPB = 256;  // 8 waves of 32 on CDNA5

    dim3 ginit((N + TPB - 1) / TPB, BATCH);
    eik_init_kernel<<<ginit, TPB, 0, stream>>>(y, yt, gn, deg, N);

    eik_degree_kernel<<<(E + TPB - 1) / TPB, TPB, 0, stream>>>(ei, ea, deg, E);

    int span = TILE * CPW;
    eik_edge_kernel<<<(E + span - 1) / span, TPB, 0, stream>>>(
        ei, ea, deg, yt, (unsigned int*)gn, alpha, E);

    eik_final_kernel<<<ginit, TPB, 0, stream>>>(gn, mask, out, N);
}